// imgs4dto3d_68968584839577
// MI455X (gfx1250) — compile-verified
//
#include <hip/hip_runtime.h>

// Problem constants (match reference)
#define CANVAS_W 200
#define HALF     15
#define PATCH_H  31
#define PATCH_W  31
#define PATCH_N  (PATCH_H * PATCH_W)   // 961
#define NEMIT    256                   // E

// ---------------------------------------------------------------------------
// Scatter kernel: one 256-thread block per (batch, emitter); scatter-add the
// 31x31 patch into the per-batch 200x200 canvas with relaxed agent-scope f32
// atomics (native no-return global_atomic_add_f32, verified via vmem count).
//
// The 20 MB output is L2-resident (192 MB L2). Patch loads are marked
// non-temporal so the 126 MB one-shot input stream does not evict the canvas
// from L2 (keeping the atomic RMWs off HBM).
//
// Element mapping over 256 threads: i = t, t+256, t+512 always valid
// (max 767 < 961); i = t+768 valid only for t < 193.
// ---------------------------------------------------------------------------
__global__ __launch_bounds__(256, 8)
void imgs4d_scatter_kernel(const float* __restrict__ img,
                           const int*   __restrict__ xyz,
                           float*       __restrict__ out) {
    const int be = blockIdx.x;       // b*256 + e
    const int b  = be >> 8;          // NEMIT == 256
    const int t  = threadIdx.x;

    // Block-uniform emitter center (rows from x, cols from y) -> s_load_b64.
    const int x = xyz[be * 3 + 0];
    const int y = xyz[be * 3 + 1];

    const float* patch = img + (size_t)be * PATCH_N;

    // gfx1250 prefetch: pull the 3844-byte patch toward the caches.
    // 31 lanes, one per 128B cacheline (last touch = float index 960 < 961).
    if (t < 31) {
        __builtin_prefetch(patch + t * 32, 0, 3);
    }

    // Top-left corner of the patch inside this batch's canvas.
    float* cbase = out + (size_t)b * (CANVAS_W * CANVAS_W)
                       + (size_t)(x - HALF) * CANVAS_W + (y - HALF);

    const int i0 = t;
    const int i1 = t + 256;
    const int i2 = t + 512;
    const int i3 = t + 768;

    // Three unconditional coalesced loads, clause-grouped, non-temporal so the
    // streaming input does not displace the L2-resident canvas.
    const float v0 = __builtin_nontemporal_load(patch + i0);
    const float v1 = __builtin_nontemporal_load(patch + i1);
    const float v2 = __builtin_nontemporal_load(patch + i2);

    // Row/col decomposition (mul-hi strength reduction for /31).
    const int r0 = i0 / PATCH_W, c0 = i0 - r0 * PATCH_W;
    const int r1 = i1 / PATCH_W, c1 = i1 - r1 * PATCH_W;
    const int r2 = i2 / PATCH_W, c2 = i2 - r2 * PATCH_W;

    // No-return relaxed agent-scope fadd -> global_atomic_add_f32 (L2 RMW).
    (void)__hip_atomic_fetch_add(cbase + r0 * CANVAS_W + c0, v0,
                                 __ATOMIC_RELAXED, __HIP_MEMORY_SCOPE_AGENT);
    (void)__hip_atomic_fetch_add(cbase + r1 * CANVAS_W + c1, v1,
                                 __ATOMIC_RELAXED, __HIP_MEMORY_SCOPE_AGENT);
    (void)__hip_atomic_fetch_add(cbase + r2 * CANVAS_W + c2, v2,
                                 __ATOMIC_RELAXED, __HIP_MEMORY_SCOPE_AGENT);

    if (i3 < PATCH_N) {
        const float v3 = __builtin_nontemporal_load(patch + i3);
        const int r3 = i3 / PATCH_W, c3 = i3 - r3 * PATCH_W;
        (void)__hip_atomic_fetch_add(cbase + r3 * CANVAS_W + c3, v3,
                                     __ATOMIC_RELAXED, __HIP_MEMORY_SCOPE_AGENT);
    }
}

// ---------------------------------------------------------------------------
// Zero the output canvas (harness poisons d_out with 0xAA).
// One global_store_b128 per thread with scale_offset; RT hint keeps the
// canvas lines resident in L2 for the scatter pass that follows.
// ---------------------------------------------------------------------------
__global__ void imgs4d_zero_kernel(float4* __restrict__ out, int n4) {
    int i = blockIdx.x * blockDim.x + threadIdx.x;
    if (i < n4) {
        out[i] = make_float4(0.f, 0.f, 0.f, 0.f);
    }
}

// ---------------------------------------------------------------------------
// Launcher
//   d_in[0] = images4D  float32 [B, E, 31, 31]
//   d_in[1] = xyz       int32   [B, E, 3]
//   d_out   = float32   [B, 1, 200, 200]
// ---------------------------------------------------------------------------
extern "C" void kernel_launch(void* const* d_in, const int* in_sizes, int n_in,
                              void* d_out, int out_size, void* d_ws, size_t ws_size,
                              hipStream_t stream) {
    const float* img = (const float*)d_in[0];
    const int*   xyz = (const int*)d_in[1];
    float*       out = (float*)d_out;

    const int B = in_sizes[1] / (NEMIT * 3);   // 128

    // Pass 1: zero the canvas (out_size = B*200*200, divisible by 4).
    const int n4 = out_size / 4;
    imgs4d_zero_kernel<<<(n4 + 255) / 256, 256, 0, stream>>>((float4*)out, n4);

    // Pass 2: one block per (batch, emitter) = 32768 blocks of 256 threads.
    imgs4d_scatter_kernel<<<B * NEMIT, 256, 0, stream>>>(img, xyz, out);
}